// GraphSAGE_86354612453978
// MI455X (gfx1250) — compile-verified
//
#include <hip/hip_runtime.h>
#include <hip/hip_bf16.h>

typedef __attribute__((ext_vector_type(2))) float v2f;
typedef __attribute__((ext_vector_type(8))) float v8f;

#define D 128               // feature width (IN_FEATS == H_FEATS == 128)
#define WAVE 32
#define LSTRIDE 132         // LDS row stride (floats): pad 128->132 (bank-conflict free, 16B aligned)

// ---------------------------------------------------------------------------
// zero a float buffer (grid-stride)
// ---------------------------------------------------------------------------
__global__ void sage_zero_f32(float* __restrict__ p, long n) {
    long i = (long)blockIdx.x * blockDim.x + threadIdx.x;
    long stride = (long)gridDim.x * blockDim.x;
    for (; i < n; i += stride) p[i] = 0.0f;
}

// ---------------------------------------------------------------------------
// degree: deg[dst[e]] += 1
// ---------------------------------------------------------------------------
__global__ void sage_degree(const int* __restrict__ dst, float* __restrict__ deg, int E) {
    int e = blockIdx.x * blockDim.x + threadIdx.x;
    if (e < E) atomicAdd(&deg[dst[e]], 1.0f);
}

// ---------------------------------------------------------------------------
// deg -> 1/max(deg,1)
// ---------------------------------------------------------------------------
__global__ void sage_recip(float* __restrict__ deg, int n) {
    int i = blockIdx.x * blockDim.x + threadIdx.x;
    if (i < n) deg[i] = 1.0f / fmaxf(deg[i], 1.0f);
}

// ---------------------------------------------------------------------------
// transpose 128x128 weight: Wt[n*128+k] = W[k*128+n]
// ---------------------------------------------------------------------------
__global__ void sage_transpose_w(const float* __restrict__ W, float* __restrict__ Wt) {
    int t = blockIdx.x * blockDim.x + threadIdx.x;   // 16384 threads
    int k = t >> 7;
    int n = t & 127;
    Wt[n * D + k] = W[k * D + n];
}

// ---------------------------------------------------------------------------
// edge scatter: one wave per edge; lane handles 4 channels (float4)
// agg[dst[e]][:] += h[src[e]][:]
// ---------------------------------------------------------------------------
__global__ void sage_scatter(const float* __restrict__ h,
                             const int* __restrict__ src,
                             const int* __restrict__ dst,
                             float* __restrict__ agg, int E) {
    int gid = blockIdx.x * blockDim.x + threadIdx.x;
    int e = gid >> 5;
    if (e >= E) return;
    int lane = threadIdx.x & (WAVE - 1);
    int s = src[e];
    int d = dst[e];
    const float4* hp = (const float4*)(h + (long)s * D);
    float4 v = hp[lane];
    float* ap = agg + (long)d * D + lane * 4;
    atomicAdd(ap + 0, v.x);
    atomicAdd(ap + 1, v.y);
    atomicAdd(ap + 2, v.z);
    atomicAdd(ap + 3, v.w);
}

// ---------------------------------------------------------------------------
// Fused SAGE linear layer using V_WMMA_F32_16X16X4_F32:
//   out = [h @ Wself + (agg * deginv) @ Wneigh + b]  (optional ReLU)
// Block = 256 threads = 8 waves; each wave computes a 16(node) x 128(col) tile.
// Both transposed weight matrices ([n][k] layout) are staged in LDS once per
// block (2 x 128 x 132 floats = 135KB of the WGP's 320KB), so the K-loop's
// B-fragments are conflict-free ds_load_b64 instead of WGP$-thrashing vmem.
// ---------------------------------------------------------------------------
__global__ void __launch_bounds__(256)
sage_gemm_wmma(const float* __restrict__ h,
               const float* __restrict__ agg,
               const float* __restrict__ deginv,
               const float* __restrict__ WtS,
               const float* __restrict__ WtN,
               const float* __restrict__ bias,
               float* __restrict__ out,
               int nrows, int do_relu) {
    __shared__ float ldsS[D * LSTRIDE];
    __shared__ float ldsN[D * LSTRIDE];

    // ---- cooperative LDS fill (all 256 threads; float4 wide, coalesced) ----
    {
        const int t = threadIdx.x;
#pragma unroll
        for (int i = 0; i < 16; ++i) {                 // 16 x 256 = 4096 float4 = 16384 floats
            const int idx4 = t + i * 256;              // float4 index within 128x128
            const int n  = idx4 >> 5;                  // row (output col)
            const int k4 = idx4 & 31;                  // float4 within row
            const float4 vs = ((const float4*)(WtS + (long)n * D))[k4];
            const float4 vn = ((const float4*)(WtN + (long)n * D))[k4];
            *(float4*)&ldsS[n * LSTRIDE + k4 * 4] = vs;
            *(float4*)&ldsN[n * LSTRIDE + k4 * 4] = vn;
        }
    }
    __syncthreads();                                   // all waves reach this before any exit

    const int lane = threadIdx.x & (WAVE - 1);
    const int wave = threadIdx.x >> 5;
    const int m0 = (blockIdx.x * 8 + wave) * 16;       // first node row of tile
    if (m0 >= nrows) return;                           // wave-uniform exit (after barrier)

    const int l15     = lane & 15;
    const int halfsel = lane >> 4;                     // 0: K=+0/+1, 1: K=+2/+3
    const int koff    = halfsel * 2;
    const int rowA    = m0 + l15;                      // A-fragment source row
    const bool rowOK  = rowA < nrows;
    const float dinv  = rowOK ? deginv[rowA] : 0.0f;

    v8f acc[8] = {};                                   // 16x128 f32 tile

    for (int k0 = 0; k0 < D; k0 += 4) {
        // A fragments (16x4 f32): lanes 0-15 -> K=k0+0/1, lanes 16-31 -> K=k0+2/3
        v2f aS = {0.0f, 0.0f};
        v2f aN = {0.0f, 0.0f};
        if (rowOK) {
            const float* hp = h   + (long)rowA * D + k0 + koff;
            const float* ap = agg + (long)rowA * D + k0 + koff;
            aS.x = hp[0];
            aS.y = hp[1];
            aN.x = ap[0] * dinv;                       // fold mean-normalization
            aN.y = ap[1] * dinv;
        }
#pragma unroll
        for (int nt = 0; nt < 8; ++nt) {
            const int col = nt * 16 + l15;
            const float* ws = &ldsS[col * LSTRIDE + k0 + koff];
            const float* wn = &ldsN[col * LSTRIDE + k0 + koff];
            v2f bS, bN;
            bS.x = ws[0]; bS.y = ws[1];
            bN.x = wn[0]; bN.y = wn[1];
            acc[nt] = __builtin_amdgcn_wmma_f32_16x16x4_f32(
                false, aS, false, bS, (short)0, acc[nt], false, false);
            acc[nt] = __builtin_amdgcn_wmma_f32_16x16x4_f32(
                false, aN, false, bN, (short)0, acc[nt], false, false);
        }
    }

    // Epilogue: C/D layout -> VGPR r: lanes0-15 = row m0+r, lanes16-31 = row m0+r+8
#pragma unroll
    for (int nt = 0; nt < 8; ++nt) {
        const int col = nt * 16 + l15;
        const float b = bias[col];
#pragma unroll
        for (int r = 0; r < 8; ++r) {
            const int row = m0 + r + halfsel * 8;
            if (row < nrows) {
                float v = acc[nt][r] + b;
                if (do_relu) v = fmaxf(v, 0.0f);
                out[(long)row * D + col] = v;
            }
        }
    }
}

// ---------------------------------------------------------------------------
extern "C" void kernel_launch(void* const* d_in, const int* in_sizes, int n_in,
                              void* d_out, int out_size, void* d_ws, size_t ws_size,
                              hipStream_t stream) {
    const float* in_feat = (const float*)d_in[0];
    const int*   src     = (const int*)d_in[1];
    const int*   dst     = (const int*)d_in[2];
    const float* Wself1  = (const float*)d_in[3];
    const float* Wneigh1 = (const float*)d_in[4];
    const float* b1      = (const float*)d_in[5];
    const float* Wself2  = (const float*)d_in[6];
    const float* Wneigh2 = (const float*)d_in[7];
    const float* b2      = (const float*)d_in[8];
    float*       out     = (float*)d_out;

    const int N = in_sizes[0] / D;     // 40000
    const int E = in_sizes[1];         // 640000

    // scratch layout (floats)
    char* ws = (char*)d_ws;
    float* agg  = (float*)ws;                          ws += (size_t)N * D * 4;
    float* h1   = (float*)ws;                          ws += (size_t)N * D * 4;
    float* deg  = (float*)ws;                          ws += ((size_t)N * 4 + 255) & ~(size_t)255;
    float* WtS1 = (float*)ws;                          ws += (size_t)D * D * 4;
    float* WtN1 = (float*)ws;                          ws += (size_t)D * D * 4;
    float* WtS2 = (float*)ws;                          ws += (size_t)D * D * 4;
    float* WtN2 = (float*)ws;                          ws += (size_t)D * D * 4;

    const long aggN = (long)N * D;
    const int ZB = 2048, ZT = 256;
    const int edge_blocks   = (E * WAVE + 255) / 256;
    const int gemm_blocks   = (N + 127) / 128;
    const int tw_blocks     = (D * D + 255) / 256;

    // ---- shared prep ----
    sage_zero_f32<<<ZB, ZT, 0, stream>>>(agg, aggN);
    sage_zero_f32<<<(N + 255) / 256, 256, 0, stream>>>(deg, N);
    sage_degree<<<(E + 255) / 256, 256, 0, stream>>>(dst, deg, E);
    sage_transpose_w<<<tw_blocks, 256, 0, stream>>>(Wself1, WtS1);
    sage_transpose_w<<<tw_blocks, 256, 0, stream>>>(Wneigh1, WtN1);
    sage_transpose_w<<<tw_blocks, 256, 0, stream>>>(Wself2, WtS2);
    sage_transpose_w<<<tw_blocks, 256, 0, stream>>>(Wneigh2, WtN2);
    sage_recip<<<(N + 255) / 256, 256, 0, stream>>>(deg, N);

    // ---- layer 1 ----
    sage_scatter<<<edge_blocks, 256, 0, stream>>>(in_feat, src, dst, agg, E);
    sage_gemm_wmma<<<gemm_blocks, 256, 0, stream>>>(in_feat, agg, deg,
                                                    WtS1, WtN1, b1, h1, N, 1);

    // ---- layer 2 ----
    sage_zero_f32<<<ZB, ZT, 0, stream>>>(agg, aggN);
    sage_scatter<<<edge_blocks, 256, 0, stream>>>(h1, src, dst, agg, E);
    sage_gemm_wmma<<<gemm_blocks, 256, 0, stream>>>(h1, agg, deg,
                                                    WtS2, WtN2, b2, out, N, 0);
}